// GraphDotProductDecoder_25623774888164
// MI455X (gfx1250) — compile-verified
//
#include <hip/hip_runtime.h>

// CDNA5 / gfx1250: per-edge dot product via V_WMMA_F32_16X16X4_F32.
// One wave32 processes a tile of 16 edges; 32 accumulating fp32 WMMAs
// (K=128 in chunks of 4) produce a 16x16 Gram tile whose diagonal is the
// 16 edge dot products. Exact fp32 math (matches fp32 reference path).

typedef __attribute__((ext_vector_type(2))) float v2f;
typedef __attribute__((ext_vector_type(8))) float v8f;

#define FEAT_D 128

__global__ __launch_bounds__(256) void edge_dot_wmma_f32(
    const float* __restrict__ h,
    const long long* __restrict__ src,
    const long long* __restrict__ dst,
    float* __restrict__ out,
    int E)
{
    const int lane = threadIdx.x & 31;
    const int waveInBlock = threadIdx.x >> 5;
    const long long tile =
        (long long)blockIdx.x * (long long)(blockDim.x >> 5) + waveInBlock;
    const long long ntiles = ((long long)E + 15) >> 4;
    if (tile >= ntiles) return;  // uniform per wave: EXEC stays all-ones

    // Edge slot handled by this lane (lanes L and L+16 share a slot and
    // cover K pairs (k0,k0+1) and (k0+2,k0+3) respectively, per the ISA's
    // 16x4 fp32 A-matrix VGPR layout).
    const int slot  = lane & 15;
    const int khalf = (lane >> 4) << 1;  // 0 or 2

    long long e = tile * 16 + slot;
    const bool valid = (e < E);
    const long long ec = valid ? e : (long long)(E - 1);  // clamp for tail

    const long long si = src[ec];
    const long long di = dst[ec];
    const float* pa = h + si * FEAT_D + khalf;  // A: hu rows   (8B aligned)
    const float* pb = h + di * FEAT_D + khalf;  // B: hv^T cols (8B aligned)

    v8f c0 = {};
    v8f c1 = {};

    // K = 128 in chunks of 4; two accumulators to relax the WMMA->WMMA
    // RAW dependence (compiler inserts any required coexec NOPs).
#pragma unroll
    for (int k0 = 0; k0 < FEAT_D; k0 += 8) {
        v2f a0 = *(const v2f*)(pa + k0);
        v2f b0 = *(const v2f*)(pb + k0);
        v2f a1 = *(const v2f*)(pa + k0 + 4);
        v2f b1 = *(const v2f*)(pb + k0 + 4);
        // 8 args: (neg_a, A, neg_b, B, c_mod, C, reuse_a, reuse_b)
        c0 = __builtin_amdgcn_wmma_f32_16x16x4_f32(
            false, a0, false, b0, (short)0, c0, false, false);
        c1 = __builtin_amdgcn_wmma_f32_16x16x4_f32(
            false, a1, false, b1, (short)0, c1, false, false);
    }

    // C/D layout: VGPR r -> row M=r (lanes 0-15) / M=r+8 (lanes 16-31), N=lane&15.
    // Diagonal (i,i): i<8 -> VGPR i @ lane i;  i>=8 -> VGPR i-8 @ lane i+16.
    // So lanes 0-7 own diag[0..7] at c[lane]; lanes 24-31 own diag[8..15]
    // at c[lane-24]. Select c[s] with a cndmask tree (s = lane & 7).
    v8f cs = c0 + c1;
    const int s = lane & 7;
    float lo01 = (s & 1) ? cs[1] : cs[0];
    float lo23 = (s & 1) ? cs[3] : cs[2];
    float hi01 = (s & 1) ? cs[5] : cs[4];
    float hi23 = (s & 1) ? cs[7] : cs[6];
    float lo = (s & 2) ? lo23 : lo01;
    float hi = (s & 2) ? hi23 : hi01;
    float diag = (s & 4) ? hi : lo;

    int i = -1;
    if (lane < 8)                    i = lane;        // diag rows 0..7
    else if (lane >= 16 && lane < 24) i = lane - 8;   // diag rows 8..15
    if (i >= 0) {
        long long eo = tile * 16 + i;
        if (eo < (long long)E) out[eo] = diag;
    }
}

extern "C" void kernel_launch(void* const* d_in, const int* in_sizes, int n_in,
                              void* d_out, int out_size, void* d_ws, size_t ws_size,
                              hipStream_t stream) {
    const float*     h   = (const float*)d_in[0];
    const long long* src = (const long long*)d_in[1];
    const long long* dst = (const long long*)d_in[2];
    float* out = (float*)d_out;

    const int E = in_sizes[1];  // number of edges
    const long long ntiles = ((long long)E + 15) / 16;
    const int wavesPerBlock = 8;  // 256 threads = 8 wave32
    const int blocks = (int)((ntiles + wavesPerBlock - 1) / wavesPerBlock);

    edge_dot_wmma_f32<<<blocks, wavesPerBlock * 32, 0, stream>>>(h, src, dst, out, E);
}